// ImportancePropagationLayer_71459665871600
// MI455X (gfx1250) — compile-verified
//
#include <hip/hip_runtime.h>
#include <hip/hip_bf16.h>

// CDNA5 / gfx1250, wave32. fp32 WMMA: D(16x16,f32) = A(16x4,f32) x B(4x16,f32) + C
typedef __attribute__((ext_vector_type(2))) float v2f;
typedef __attribute__((ext_vector_type(8))) float v8f;

#define WMMA_F32X4(a, b, c) \
  __builtin_amdgcn_wmma_f32_16x16x4_f32(false, (a), false, (b), (short)0, (c), false, false)

#define DD 128          // feature dim (D_IN == D_OUT)
#define WPB 8           // waves per block (256 threads)

// Interleaved B layout: Wp[(k>>1)*256 + col*2 + (k&1)] = W[k*128 + col]
// -> per-lane B pair (W[k][col], W[k+1][col]) is one contiguous b64 load.

// ---------------------------------------------------------------------------
// Kernel 0: repack W_upd (256x128) and W_gate rows 0..127 into B-pair layout.
// ---------------------------------------------------------------------------
__global__ __launch_bounds__(256) void k_repack(
    const float* __restrict__ Wu, const float* __restrict__ Wg,
    float* __restrict__ Wup, float* __restrict__ Wgp)
{
  const int i = blockIdx.x * 256 + threadIdx.x;
  if (i < 2 * DD * DD) { // 32768 W_upd elements
    const int k = i >> 7, col = i & 127;
    Wup[(k >> 1) * 256 + col * 2 + (k & 1)] = Wu[i];
  }
  if (i < DD * DD) {     // 16384 W_gate elements (rows 0..127)
    const int k = i >> 7, col = i & 127;
    Wgp[(k >> 1) * 256 + col * 2 + (k & 1)] = Wg[i];
  }
}

// ---------------------------------------------------------------------------
// Kernel 1: m = x @ W_msg + b_msg   (and zero the aggr scratch buffer)
// One wave computes a 16x128 output strip. W_msg staged *interleaved* in LDS.
// A layout (16x4 f32): lane 0-15 -> row=lane, K=k0..k0+1 ; lane 16-31 -> K=k0+2..k0+3
// B layout (4x16 f32): lane 0-15 -> col=lane, K=k0..k0+1 ; lane 16-31 -> K=k0+2..k0+3
// C layout: VGPR r -> row r (lanes 0-15) / row r+8 (lanes 16-31), col = lane&15
// ---------------------------------------------------------------------------
__global__ __launch_bounds__(256) void k_msg_gemm(
    const float* __restrict__ x, const float* __restrict__ Wm,
    const float* __restrict__ bm, float* __restrict__ m,
    float* __restrict__ aggr, int nStrips)
{
  __shared__ float sW[DD * DD]; // 64 KB, interleaved B-pair layout
  for (int i = threadIdx.x; i < DD * DD; i += 256) {
    const int k = i >> 7, col = i & 127;
    sW[(k >> 1) * 256 + col * 2 + (k & 1)] = Wm[i];
  }
  __syncthreads();

  const int wave = blockIdx.x * WPB + (threadIdx.x >> 5);
  if (wave >= nStrips) return;
  const int lane = threadIdx.x & 31;
  const int lo = lane & 15, hi = lane >> 4;
  const long row0 = (long)wave * 16;
  const long rowA = row0 + lo;

  v8f acc[8];
#pragma unroll
  for (int nt = 0; nt < 8; ++nt) {
    float bv = bm[nt * 16 + lo];
    acc[nt] = (v8f){bv, bv, bv, bv, bv, bv, bv, bv};
  }

  for (int k0 = 0; k0 < DD; k0 += 4) {
    const int kk = k0 + (hi << 1);
    v2f a = *(const v2f*)(x + rowA * DD + kk);
    const float* bp = sW + ((k0 >> 1) + hi) * 256 + lo * 2;
#pragma unroll
    for (int nt = 0; nt < 8; ++nt) {
      v2f b = *(const v2f*)(bp + nt * 32);   // single ds_load_b64 per tile
      acc[nt] = WMMA_F32X4(a, b, acc[nt]);
    }
  }

#pragma unroll
  for (int nt = 0; nt < 8; ++nt) {
    const int col = nt * 16 + lo;
#pragma unroll
    for (int r = 0; r < 8; ++r) {
      const long row = row0 + r + hi * 8;
      m[row * DD + col]    = acc[nt][r];
      aggr[row * DD + col] = 0.0f;   // zero scatter target every call
    }
  }
}

// ---------------------------------------------------------------------------
// Kernel 2: aggr[dst] += m[src]  — one wave per edge, lane handles 4 floats.
// ---------------------------------------------------------------------------
__global__ __launch_bounds__(256) void k_scatter(
    const float* __restrict__ m, const int* __restrict__ ei,
    float* __restrict__ aggr, int E)
{
  const int e = blockIdx.x * WPB + (threadIdx.x >> 5);
  if (e >= E) return;
  const int lane = threadIdx.x & 31;
  const int src = ei[e];
  const int dst = ei[E + e];
  const float4 v = *(const float4*)(m + (long)src * DD + lane * 4);
  float* p = aggr + (long)dst * DD + lane * 4;
  unsafeAtomicAdd(p + 0, v.x);
  unsafeAtomicAdd(p + 1, v.y);
  unsafeAtomicAdd(p + 2, v.z);
  unsafeAtomicAdd(p + 3, v.w);
}

// ---------------------------------------------------------------------------
// Kernel 3 (fused): conv = [x|aggr]@W_upd + b_upd
//                   gate = sigmoid([conv|imp]@W_gate + b_gate)
//                   out  = gate*conv + (1-gate)*x
//                   pimp = out @ W_imp + b_imp
// One wave per 16-row strip; conv staged per-wave in LDS to re-feed as WMMA A.
// B operands come from the repacked (pair-interleaved) weights in d_ws:
// one coalesced global_load_b64 per tile per k-step.
// ---------------------------------------------------------------------------
__global__ __launch_bounds__(256) void k_update(
    const float* __restrict__ x, const float* __restrict__ aggr,
    const float* __restrict__ imp,
    const float* __restrict__ Wup, const float* __restrict__ bu,
    const float* __restrict__ Wgp, const float* __restrict__ Wg,
    const float* __restrict__ bg,
    const float* __restrict__ Wi, const float* __restrict__ bi,
    float* __restrict__ out, float* __restrict__ pimp, int nStrips)
{
  __shared__ float sC[WPB * 16 * DD]; // 64KB: per-wave 16x128 conv staging
  const int wslot = threadIdx.x >> 5;
  const int wave = blockIdx.x * WPB + wslot;
  if (wave >= nStrips) return;
  const int lane = threadIdx.x & 31;
  const int lo = lane & 15, hi = lane >> 4;
  const long row0 = (long)wave * 16;
  const long rowA = row0 + lo;
  float* myC = sC + wslot * (16 * DD);

  // ---- conv = [x | aggr] @ W_upd + b_upd  (K = 256) ----
  v8f conv[8];
#pragma unroll
  for (int nt = 0; nt < 8; ++nt) {
    float bv = bu[nt * 16 + lo];
    conv[nt] = (v8f){bv, bv, bv, bv, bv, bv, bv, bv};
  }
  for (int k0 = 0; k0 < 2 * DD; k0 += 4) {
    const int kk = k0 + (hi << 1);
    const float* asrc = (kk < DD) ? (x + rowA * DD + kk)
                                  : (aggr + rowA * DD + (kk - DD));
    v2f a = *(const v2f*)asrc;
    const float* bp = Wup + ((k0 >> 1) + hi) * 256 + lo * 2;
#pragma unroll
    for (int nt = 0; nt < 8; ++nt) {
      v2f b = *(const v2f*)(bp + nt * 32);
      conv[nt] = WMMA_F32X4(a, b, conv[nt]);
    }
  }

  // stage conv into LDS row-major [16][128] (own region only, no barrier)
#pragma unroll
  for (int nt = 0; nt < 8; ++nt)
#pragma unroll
    for (int r = 0; r < 8; ++r)
      myC[(r + hi * 8) * DD + nt * 16 + lo] = conv[nt][r];

  // ---- gate = sigmoid([conv | imp] @ W_gate + b_gate)  (K = 129) ----
  v8f g[8];
#pragma unroll
  for (int nt = 0; nt < 8; ++nt) {
    float bv = bg[nt * 16 + lo];
    g[nt] = (v8f){bv, bv, bv, bv, bv, bv, bv, bv};
  }
  for (int k0 = 0; k0 < DD; k0 += 4) {
    const int kk = k0 + (hi << 1);
    v2f a = *(const v2f*)(myC + lo * DD + kk);   // conv as A, from LDS
    const float* bp = Wgp + ((k0 >> 1) + hi) * 256 + lo * 2;
#pragma unroll
    for (int nt = 0; nt < 8; ++nt) {
      v2f b = *(const v2f*)(bp + nt * 32);
      g[nt] = WMMA_F32X4(a, b, g[nt]);
    }
  }
  // rank-1 importance column (W_gate row 128) + sigmoid
  float impr[8];
#pragma unroll
  for (int r = 0; r < 8; ++r) impr[r] = imp[row0 + r + hi * 8];
#pragma unroll
  for (int nt = 0; nt < 8; ++nt) {
    const float wl = Wg[DD * DD + nt * 16 + lo];
#pragma unroll
    for (int r = 0; r < 8; ++r) {
      float z = g[nt][r] + impr[r] * wl;
      g[nt][r] = 1.0f / (1.0f + __expf(-z));
    }
  }

  // ---- out = g*conv + (1-g)*x ; pimp = out @ W_imp + b_imp ----
  const float bimp = bi[0];
  float pacc[8] = {0, 0, 0, 0, 0, 0, 0, 0};
#pragma unroll
  for (int nt = 0; nt < 8; ++nt) {
    const int col = nt * 16 + lo;
    const float wi = Wi[col];
#pragma unroll
    for (int r = 0; r < 8; ++r) {
      const long row = row0 + r + hi * 8;
      const float xv = x[row * DD + col];
      const float gv = g[nt][r];
      const float o = gv * conv[nt][r] + (1.0f - gv) * xv;
      out[row * DD + col] = o;
      pacc[r] += o * wi;
    }
  }
  // reduce each row's partial across the 16 lanes of its half-wave
#pragma unroll
  for (int r = 0; r < 8; ++r) {
    float p = pacc[r];
    p += __shfl_xor(p, 1, 32);
    p += __shfl_xor(p, 2, 32);
    p += __shfl_xor(p, 4, 32);
    p += __shfl_xor(p, 8, 32);
    if (lo == 0) pimp[row0 + r + hi * 8] = p + bimp;
  }
}

// ---------------------------------------------------------------------------
extern "C" void kernel_launch(void* const* d_in, const int* in_sizes, int n_in,
                              void* d_out, int out_size, void* d_ws, size_t ws_size,
                              hipStream_t stream) {
  const float* x   = (const float*)d_in[0];
  const int*   ei  = (const int*)d_in[1];     // edge_index [2,E]
  const float* imp = (const float*)d_in[2];
  const float* Wm  = (const float*)d_in[3];
  const float* bm  = (const float*)d_in[4];
  const float* Wu  = (const float*)d_in[5];
  const float* bu  = (const float*)d_in[6];
  const float* Wg  = (const float*)d_in[7];
  const float* bg  = (const float*)d_in[8];
  const float* Wi  = (const float*)d_in[9];
  const float* bi  = (const float*)d_in[10];

  const int N = in_sizes[0] / DD;   // 50000
  const int E = in_sizes[1] / 2;    // 800000

  float* out  = (float*)d_out;
  float* pimp = out + (size_t)N * DD;
  float* m    = out;                // stage m_nodes in d_out (overwritten later)

  float* aggr = (float*)d_ws;               // N*128 floats of scratch
  float* Wup  = aggr + (size_t)N * DD;      // 32768 floats, repacked W_upd
  float* Wgp  = Wup + 2 * DD * DD;          // 16384 floats, repacked W_gate[0:128]

  const int nStrips = N / 16;                    // 3125 (N divisible by 16)
  const int gGemm   = (nStrips + WPB - 1) / WPB; // 391 blocks
  const int gScat   = (E + WPB - 1) / WPB;       // 100000 blocks

  k_repack  <<<(2 * DD * DD + 255) / 256, 256, 0, stream>>>(Wu, Wg, Wup, Wgp);
  k_msg_gemm<<<gGemm, 256, 0, stream>>>(x, Wm, bm, m, aggr, nStrips);
  k_scatter <<<gScat, 256, 0, stream>>>(m, ei, aggr, E);
  k_update  <<<gGemm, 256, 0, stream>>>(x, aggr, imp, Wup, bu, Wgp, Wg, bg,
                                        Wi, bi, out, pimp, nStrips);
}